// MsaHmmLayer_35124242546874
// MI455X (gfx1250) — compile-verified
//
#include <hip/hip_runtime.h>

#define Qn 1024
#define An 26
#define Bn 128
#define Ln 1024
#define ES 1028   // padded LDS row stride (floats): 4*m mod 64 distinct -> no bank conflicts

typedef __attribute__((ext_vector_type(2))) float v2f;
typedef __attribute__((ext_vector_type(8))) float v8f;

// Row softmax of trans_logits [Q,Q] -> packed layout for WMMA B-fragments:
// tpack[(k>>1)*2*Q + 2*n + (k&1)] = T[k][n], so {T[2p][n],T[2p+1][n]} is one float2.
__global__ __launch_bounds__(256) void softmax_rows_pack(const float* __restrict__ logits,
                                                         float* __restrict__ tpack) {
  __shared__ float red[256];
  const int row = blockIdx.x;
  const int j = threadIdx.x;
  const float* src = logits + (size_t)row * Qn;
  float mx = -3.402823466e+38f;
  for (int i = j; i < Qn; i += 256) mx = fmaxf(mx, src[i]);
  red[j] = mx; __syncthreads();
  for (int s = 128; s > 0; s >>= 1) { if (j < s) red[j] = fmaxf(red[j], red[j + s]); __syncthreads(); }
  const float M = red[0]; __syncthreads();
  float sum = 0.f;
  for (int i = j; i < Qn; i += 256) sum += expf(src[i] - M);
  red[j] = sum; __syncthreads();
  for (int s = 128; s > 0; s >>= 1) { if (j < s) red[j] += red[j + s]; __syncthreads(); }
  const float rs = 1.0f / red[0];
  float* dst = tpack + (size_t)(row >> 1) * (2 * Qn) + (row & 1);
  for (int i = j; i < Qn; i += 256) dst[2 * i] = expf(src[i] - M) * rs;
}

// One workgroup (1024 thr = 32 waves) owns 16 batch rows for the full sequence.
__global__ __launch_bounds__(1024) void hmm_forward(const float* __restrict__ inputs,
                                                    const float* __restrict__ init_logits,
                                                    const float* __restrict__ emis_logits,
                                                    const float* __restrict__ tpack,
                                                    float* __restrict__ loglik) {
  __shared__ float alpha[16 * ES];   // normalized alpha_{t-1}
  __shared__ float ework[16 * ES];   // e_t, then (alpha@T)*e_t in place
  __shared__ float inpt[16 * 32];    // observation tile [16][26] (padded)
  __shared__ float red[1024];
  __shared__ float sinv[16];

  const int tid  = threadIdx.x;
  const int lane = tid & 31;
  const int w    = tid >> 5;
  const int b0   = blockIdx.x * 16;

  // Per-thread emission row softmax: bm[a] = Bm[q=tid][a], lives in VGPRs all kernel.
  float bm[An];
  {
    const float* er = emis_logits + tid * An;
    float m2 = -3.402823466e+38f;
#pragma unroll
    for (int a = 0; a < An; ++a) { bm[a] = er[a]; m2 = fmaxf(m2, bm[a]); }
    float s2 = 0.f;
#pragma unroll
    for (int a = 0; a < An; ++a) { bm[a] = expf(bm[a] - m2); s2 += bm[a]; }
    const float r2 = 1.0f / s2;
#pragma unroll
    for (int a = 0; a < An; ++a) bm[a] *= r2;
  }

  // pi = softmax(init_logits); pi_q held per-thread (q = tid). Use ework as scratch.
  float pi_q;
  {
    float* pr = ework;
    const float x = init_logits[tid];
    pr[tid] = x; __syncthreads();
    for (int s = 512; s > 0; s >>= 1) { if (tid < s) pr[tid] = fmaxf(pr[tid], pr[tid + s]); __syncthreads(); }
    const float M = pr[0]; __syncthreads();
    const float e = expf(x - M);
    pr[tid] = e; __syncthreads();
    for (int s = 512; s > 0; s >>= 1) { if (tid < s) pr[tid] += pr[tid + s]; __syncthreads(); }
    pi_q = e / pr[0];
    __syncthreads();
  }

  const v2f* __restrict__ tp2 = (const v2f*)tpack;
  float ll = 0.f;  // meaningful for tid < 16 (row = tid)

  for (int t = 0; t < Ln; ++t) {
    // Stage inputs[b0..b0+15][t][0..25] into LDS.
    if (tid < 16 * An) {
      const int m = tid / An, a = tid - m * An;
      inpt[m * 32 + a] = inputs[((size_t)(b0 + m) * Ln + t) * An + a];
    }
    __syncthreads();

    // e_t[m][q=tid] = sum_a inpt[m][a] * Bm[q][a]; fold pi at t==0.
#pragma unroll 4
    for (int m = 0; m < 16; ++m) {
      float s = 0.f;
#pragma unroll
      for (int a = 0; a < An; ++a) s = fmaf(inpt[m * 32 + a], bm[a], s);
      ework[m * ES + tid] = (t == 0) ? s * pi_q : s;
    }
    __syncthreads();

    if (t > 0) {
      // (alpha @ T) via fp32 WMMA 16x16x4.
      // Wave w owns N-tiles {w, w+32}: fused K-loop, one shared A-fragment,
      // two independent accumulator chains for XDL pipelining.
      const int hi  = lane >> 4;     // selects K pair / upper M half per ISA layout
      const int ml  = lane & 15;
      const int nA  = (w << 4) + ml;          // column in tile 0
      const int nB  = ((w + 32) << 4) + ml;   // column in tile 1
      v8f acc0 = {};
      v8f acc1 = {};
      for (int kk = 0; kk < Qn; kk += 4) {
        const int k0 = kk + 2 * hi;
        const size_t prow = (size_t)(k0 >> 1) * Qn;
        v2f af  = *(const v2f*)&alpha[ml * ES + k0];   // A[m][k0..k0+1] (one ds_load_b64)
        v2f bf0 = tp2[prow + nA];                      // {T[k0][nA],T[k0+1][nA]}
        v2f bf1 = tp2[prow + nB];
        acc0 = __builtin_amdgcn_wmma_f32_16x16x4_f32(false, af, false, bf0,
                                                     (short)0, acc0, false, false);
        acc1 = __builtin_amdgcn_wmma_f32_16x16x4_f32(false, af, false, bf1,
                                                     (short)0, acc1, false, false);
      }
      // Epilogue: multiply by e_t, write back in place (each (m,n) owned by one lane).
#pragma unroll
      for (int r = 0; r < 8; ++r) {
        const int m = r + 8 * hi;
        ework[m * ES + nA] = acc0[r] * ework[m * ES + nA];
      }
#pragma unroll
      for (int r = 0; r < 8; ++r) {
        const int m = r + 8 * hi;
        ework[m * ES + nB] = acc1[r] * ework[m * ES + nB];
      }
      __syncthreads();
    }

    // Row sums s[m] = sum_q ework[m][q]
    {
      const int row = tid >> 6, j = tid & 63;
      float p = 0.f;
#pragma unroll
      for (int i = 0; i < 16; ++i) p += ework[row * ES + j + 64 * i];
      red[row * 64 + j] = p;
    }
    __syncthreads();
    if (tid < 16) {
      float s = 0.f;
#pragma unroll
      for (int j2 = 0; j2 < 64; ++j2) s += red[tid * 64 + j2];
      ll += logf(s);
      sinv[tid] = 1.0f / s;
    }
    __syncthreads();
    // alpha = ework / s
#pragma unroll 4
    for (int m = 0; m < 16; ++m) alpha[m * ES + tid] = ework[m * ES + tid] * sinv[m];
    __syncthreads();
  }

  if (tid < 16) loglik[b0 + tid] = ll;
}

extern "C" void kernel_launch(void* const* d_in, const int* in_sizes, int n_in,
                              void* d_out, int out_size, void* d_ws, size_t ws_size,
                              hipStream_t stream) {
  (void)in_sizes; (void)n_in; (void)out_size; (void)ws_size;
  const float* inputs       = (const float*)d_in[0];   // [B,L,A]
  const float* init_logits  = (const float*)d_in[1];   // [Q]
  const float* trans_logits = (const float*)d_in[2];   // [Q,Q]
  const float* emis_logits  = (const float*)d_in[3];   // [Q,A]
  float* out   = (float*)d_out;                        // [B]
  float* tpack = (float*)d_ws;                         // 4 MB packed transition matrix

  softmax_rows_pack<<<Qn, 256, 0, stream>>>(trans_logits, tpack);
  hmm_forward<<<Bn / 16, 1024, 0, stream>>>(inputs, init_logits, emis_logits, tpack, out);
}